// BasicGNN_36541581754794
// MI455X (gfx1250) — compile-verified
//
#include <hip/hip_runtime.h>
#include <math.h>

typedef float v2f __attribute__((ext_vector_type(2)));
typedef float v8f __attribute__((ext_vector_type(8)));

#define HDIM 128
#define LDS_STRIDE 130           // pad: lanes reading col*stride+k hit distinct banks
#define WAVES_PER_BLOCK 8

// out[row][col] = sum_k X[row][k] * W[col][k] + bias[col]  (optional ReLU)
// W + bias staged in LDS per block; one wave per 16-row tile, 16x128 strip,
// full fp32 via V_WMMA_F32_16X16X4_F32 (8 accumulators of v8f).
// A frag (16x4 f32): lanes 0-15 hold M=lane, VGPR0/1 = K0/K1; lanes 16-31 = K2/K3.
// B frag (4x16 f32): lanes 0-15 hold N=lane, VGPR0/1 = K0/K1; lanes 16-31 = K2/K3.
// C/D (16x16 f32): VGPR r -> M = r + 8*(lane>>4), N = lane&15.
__global__ void gemm_bias_kernel(const float* __restrict__ X,
                                 const float* __restrict__ W,
                                 const float* __restrict__ bias,
                                 float* __restrict__ out,
                                 int nrows, int do_relu) {
  __shared__ float Wlds[HDIM * LDS_STRIDE];   // ~65 KB of the 320 KB WGP LDS
  __shared__ float blds[HDIM];

  // Cooperative stage: 8192 float2 elements, coalesced, 8B-aligned in LDS.
  for (int i = threadIdx.x; i < (HDIM * HDIM) / 2; i += blockDim.x) {
    int c = i >> 6;
    int k = (i & 63) << 1;
    *(v2f*)&Wlds[c * LDS_STRIDE + k] = *(const v2f*)(W + (size_t)c * HDIM + k);
  }
  if (threadIdx.x < HDIM) blds[threadIdx.x] = bias[threadIdx.x];
  __syncthreads();

  int wave = threadIdx.x >> 5;
  int lane = threadIdx.x & 31;
  int tileM = blockIdx.x * WAVES_PER_BLOCK + wave;
  if (tileM * 16 < nrows) {                 // wave-uniform: EXEC stays all-ones
    int mrow = lane & 15;
    int row  = tileM * 16 + mrow;
    int koff = (lane >> 4) * 2;             // which K-pair this half-wave owns
    const float* xrow = X + (size_t)row * HDIM + koff;

    v8f zero = {};
    v8f acc[8];
#pragma unroll
    for (int j = 0; j < 8; ++j) acc[j] = zero;

#pragma unroll 4
    for (int k0 = 0; k0 < HDIM; k0 += 4) {
      v2f a = *(const v2f*)(xrow + k0);
      v2f b[8];
#pragma unroll
      for (int j = 0; j < 8; ++j)           // batch all ds loads first
        b[j] = *(const v2f*)&Wlds[(j * 16 + mrow) * LDS_STRIDE + k0 + koff];
#pragma unroll
      for (int j = 0; j < 8; ++j)           // then back-to-back WMMAs
        acc[j] = __builtin_amdgcn_wmma_f32_16x16x4_f32(
            false, a, false, b[j], (short)0, acc[j], false, false);
    }

    int mbase = tileM * 16 + ((lane >> 4) << 3);
#pragma unroll
    for (int j = 0; j < 8; ++j) {
      int col = j * 16 + mrow;
      float bv = blds[col];
#pragma unroll
      for (int r = 0; r < 8; ++r) {
        float v = acc[j][r] + bv;
        if (do_relu) v = fmaxf(v, 0.0f);
        out[(size_t)(mbase + r) * HDIM + col] = v;
      }
    }
  }
}

__global__ void zero_kernel(float* __restrict__ p, long long n) {
  long long i = (long long)blockIdx.x * blockDim.x + threadIdx.x;
  if (i < n) p[i] = 0.0f;
}

// One wave per edge. Lane l handles features {l, l+32, l+64, l+96} so each of
// the 4 atomic instructions covers a contiguous 128B region of agg[dst].
__global__ void scatter_kernel(const float* __restrict__ xw,
                               const int* __restrict__ src,
                               const int* __restrict__ dst,
                               float* __restrict__ agg,
                               float* __restrict__ cnt,
                               int E) {
  long long t = (long long)blockIdx.x * blockDim.x + threadIdx.x;
  int e = (int)(t >> 5);
  if (e >= E) return;
  int lane = (int)(t & 31);
  int s = src[e];
  int d = dst[e];
  const float* vp = xw + (size_t)s * HDIM + lane;
  float v0 = vp[0];
  float v1 = vp[32];
  float v2 = vp[64];
  float v3 = vp[96];
  float* ap = agg + (size_t)d * HDIM + lane;
  unsafeAtomicAdd(ap +  0, v0);
  unsafeAtomicAdd(ap + 32, v1);
  unsafeAtomicAdd(ap + 64, v2);
  unsafeAtomicAdd(ap + 96, v3);
  if (lane == 0) unsafeAtomicAdd(cnt + d, 1.0f);
}

// agg = relu(agg / max(cnt,1))  in-place
__global__ void mean_relu_kernel(float* __restrict__ agg,
                                 const float* __restrict__ cnt,
                                 long long total) {
  long long t = (long long)blockIdx.x * blockDim.x + threadIdx.x;
  if (t >= total) return;
  int n = (int)(t >> 7);                    // HDIM == 128
  float c = fmaxf(cnt[n], 1.0f);
  agg[t] = fmaxf(agg[t] / c, 0.0f);
}

// Segment-max over sorted batch ids. Block = 128 threads (one feature column
// each), scans a contiguous chunk of nodes, flushes per-graph partial maxima
// with integer atomicMax (valid: post-ReLU values >= 0, g initialized to 0).
#define SM_ROWS 512
__global__ void segmax_kernel(const float* __restrict__ h,
                              const int* __restrict__ batch,
                              float* __restrict__ g, int N) {
  int c = threadIdx.x;                      // 0..127
  int n0 = blockIdx.x * SM_ROWS;
  int n1 = n0 + SM_ROWS;
  if (n1 > N) n1 = N;
  if (n0 >= N) return;
  float m = 0.0f;
  int cur = batch[n0];
  for (int n = n0; n < n1; ++n) {
    int b = batch[n];
    if (b != cur) {
      atomicMax((int*)&g[(size_t)cur * HDIM + c], __float_as_int(m));
      m = 0.0f;
      cur = b;
    }
    float v = h[(size_t)n * HDIM + c];
    m = fmaxf(m, v);
  }
  atomicMax((int*)&g[(size_t)cur * HDIM + c], __float_as_int(m));
}

// z2 = z1 @ Wf2.T + bf2 ; out = log_softmax(z2) over 2 classes.
__global__ void head2_kernel(const float* __restrict__ z1,
                             const float* __restrict__ Wf2,
                             const float* __restrict__ bf2,
                             float* __restrict__ out, int G) {
  int gi = blockIdx.x * blockDim.x + threadIdx.x;
  if (gi >= G) return;
  float a0 = bf2[0], a1 = bf2[1];
  const float* zr = z1 + (size_t)gi * HDIM;
  for (int k = 0; k < HDIM; ++k) {
    float v = zr[k];
    a0 += v * Wf2[k];
    a1 += v * Wf2[HDIM + k];
  }
  float mx = fmaxf(a0, a1);
  float lse = mx + logf(expf(a0 - mx) + expf(a1 - mx));
  out[(size_t)gi * 2 + 0] = a0 - lse;
  out[(size_t)gi * 2 + 1] = a1 - lse;
}

extern "C" void kernel_launch(void* const* d_in, const int* in_sizes, int n_in,
                              void* d_out, int out_size, void* d_ws, size_t ws_size,
                              hipStream_t stream) {
  const float* x   = (const float*)d_in[0];
  const int*   ei  = (const int*)d_in[1];
  const int*   bat = (const int*)d_in[2];
  const float* W1  = (const float*)d_in[3];
  const float* b1  = (const float*)d_in[4];
  const float* W2  = (const float*)d_in[5];
  const float* b2  = (const float*)d_in[6];
  const float* Wf1 = (const float*)d_in[7];
  const float* bf1 = (const float*)d_in[8];
  const float* Wf2 = (const float*)d_in[9];
  const float* bf2 = (const float*)d_in[10];
  float* out = (float*)d_out;

  const int N = in_sizes[0] / HDIM;
  const int E = in_sizes[1] / 2;
  const int G = out_size / 2;
  const int* src = ei;
  const int* dst = ei + E;

  // workspace layout (~103 MB): xw[N*H] | agg[N*H] | cnt[N] | gbuf[G*H] | z1[G*H]
  float* xw   = (float*)d_ws;
  float* agg  = xw + (size_t)N * HDIM;
  float* cnt  = agg + (size_t)N * HDIM;
  float* gbuf = cnt + N;
  float* z1   = gbuf + (size_t)G * HDIM;

  const int tiles = (N + 15) / 16;
  const dim3 gemm_grid((tiles + WAVES_PER_BLOCK - 1) / WAVES_PER_BLOCK);
  const long long tot = (long long)N * HDIM;
  const long long sthreads = (long long)E * 32;
  const int zgrid  = (int)((tot + N + 255) / 256);
  const int sgrid  = (int)((sthreads + 255) / 256);
  const int mgrid  = (int)((tot + 255) / 256);

  // ---- layer 1: xw = x@W1.T+b1 ; scatter-mean ; relu ----
  gemm_bias_kernel<<<gemm_grid, 256, 0, stream>>>(x, W1, b1, xw, N, 0);
  zero_kernel<<<zgrid, 256, 0, stream>>>(agg, tot + N);           // agg + cnt contiguous
  scatter_kernel<<<sgrid, 256, 0, stream>>>(xw, src, dst, agg, cnt, E);
  mean_relu_kernel<<<mgrid, 256, 0, stream>>>(agg, cnt, tot);     // h1 in agg

  // ---- layer 2 ----
  gemm_bias_kernel<<<gemm_grid, 256, 0, stream>>>(agg, W2, b2, xw, N, 0);
  zero_kernel<<<zgrid, 256, 0, stream>>>(agg, tot + N);
  scatter_kernel<<<sgrid, 256, 0, stream>>>(xw, src, dst, agg, cnt, E);
  mean_relu_kernel<<<mgrid, 256, 0, stream>>>(agg, cnt, tot);     // h2 in agg

  // ---- global max pool ----
  zero_kernel<<<(G * HDIM + 255) / 256, 256, 0, stream>>>(gbuf, (long long)G * HDIM);
  segmax_kernel<<<(N + SM_ROWS - 1) / SM_ROWS, HDIM, 0, stream>>>(agg, bat, gbuf, N);

  // ---- head: z1 = relu(g@Wf1.T+bf1) ; log_softmax(z1@Wf2.T+bf2) ----
  const int htiles = (G + 15) / 16;
  gemm_bias_kernel<<<(htiles + WAVES_PER_BLOCK - 1) / WAVES_PER_BLOCK, 256, 0, stream>>>(
      gbuf, Wf1, bf1, z1, G, 1);
  head2_kernel<<<(G + 63) / 64, 64, 0, stream>>>(z1, Wf2, bf2, out, G);
}